// DgCD_25426206392550
// MI455X (gfx1250) — compile-verified
//
#include <hip/hip_runtime.h>
#include <math.h>

#define B_    128
#define C_    2048
#define HW_   196
#define ENV_  32
#define NENV_ 4

typedef __attribute__((ext_vector_type(2))) float v2f;
typedef __attribute__((ext_vector_type(8))) float v8f;

// ---------------- block reductions (deterministic, fixed order) --------------
template<int OP> // 0=sum 1=max 2=min
__device__ __forceinline__ float block_reduce(float v, float* sred) {
#pragma unroll
  for (int o = 16; o > 0; o >>= 1) {
    float t = __shfl_xor(v, o, 32);
    v = (OP == 0) ? (v + t) : (OP == 1 ? fmaxf(v, t) : fminf(v, t));
  }
  const int lane = threadIdx.x & 31;
  const int wid  = threadIdx.x >> 5;
  const int nw   = blockDim.x >> 5;
  __syncthreads();                // protect sred from prior use
  if (lane == 0) sred[wid] = v;
  __syncthreads();
  float acc = sred[0];
  for (int i = 1; i < nw; ++i) {
    float t = sred[i];
    acc = (OP == 0) ? (acc + t) : (OP == 1 ? fmaxf(acc, t) : fminf(acc, t));
  }
  return acc;
}

// ---------------- K1: avg[b,c] = mean over 196 spatial elems -----------------
// one wave per (b,c) row, consecutive lanes -> coalesced streaming of x
__global__ void k_avg(const float* __restrict__ x, float* __restrict__ avg) {
  const int row  = blockIdx.x * 8 + (threadIdx.x >> 5);
  const int lane = threadIdx.x & 31;
  const float* p = x + (size_t)row * HW_;
  float s = 0.f;
  for (int j = lane; j < HW_; j += 32) s += p[j];
#pragma unroll
  for (int o = 16; o > 0; o >>= 1) s += __shfl_xor(s, o, 32);
  if (lane == 0) avg[row] = s * (1.0f / HW_);
}

// ---------------- K2: per-channel env/total z-scores -------------------------
__global__ void k_stats(const float* __restrict__ avg,
                        float* __restrict__ ze, float* __restrict__ zt) {
  const int c = blockIdx.x * blockDim.x + threadIdx.x;  // 0..C_-1
  float em[NENV_], eiv[NENV_];
  float ts = 0.f, tss = 0.f;
#pragma unroll
  for (int e = 0; e < NENV_; ++e) {
    float s = 0.f, ss = 0.f;
    for (int i = 0; i < ENV_; ++i) {
      float a = avg[(e * ENV_ + i) * C_ + c];
      s += a; ss += a * a;
    }
    ts += s; tss += ss;
    float m = s * (1.0f / ENV_);
    float v = (ss - (float)ENV_ * m * m) * (1.0f / (ENV_ - 1));  // ddof=1
    em[e]  = m;
    eiv[e] = 1.0f / sqrtf(v + 1e-5f);
  }
  float tm  = ts * (1.0f / B_);
  float tv  = (tss - (float)B_ * tm * tm) * (1.0f / (B_ - 1));
  float tiv = 1.0f / sqrtf(tv + 1e-5f);
#pragma unroll
  for (int e = 0; e < NENV_; ++e)
    for (int i = 0; i < ENV_; ++i) {
      int b = e * ENV_ + i;
      float a = avg[b * C_ + c];
      ze[b * C_ + c] = (a - em[e]) * eiv[e];
      zt[b * C_ + c] = (a - tm) * tiv;
    }
}

// ---------------- K3: analytic KL grads + perturbed alignment ----------------
__global__ void k_align(const float* __restrict__ zep, const float* __restrict__ ztp,
                        const float* __restrict__ rho_p,
                        float* __restrict__ enva, float* __restrict__ tota) {
  __shared__ float sred[8];
  const int b   = blockIdx.x;
  const int tid = threadIdx.x;
  const float rho  = rho_p[0];
  const float invB = 1.0f / B_;
  float ze[8], zt[8];
#pragma unroll
  for (int i = 0; i < 8; ++i) {
    int c = tid + i * 256;
    ze[i] = zep[b * C_ + c];
    zt[i] = ztp[b * C_ + c];
  }
  float me = -INFINITY, mt = -INFINITY;
#pragma unroll
  for (int i = 0; i < 8; ++i) { me = fmaxf(me, ze[i]); mt = fmaxf(mt, zt[i]); }
  me = block_reduce<1>(me, sred);
  mt = block_reduce<1>(mt, sred);
  float se = 0.f, st = 0.f;
#pragma unroll
  for (int i = 0; i < 8; ++i) { se += expf(ze[i] - me); st += expf(zt[i] - mt); }
  se = block_reduce<0>(se, sred);
  st = block_reduce<0>(st, sred);
  const float lse = logf(se), lst = logf(st);
  float kl = 0.f;
#pragma unroll
  for (int i = 0; i < 8; ++i) {
    float lp = ze[i] - me - lse, lq = zt[i] - mt - lst;
    kl += expf(lp) * (lp - lq);
  }
  kl = block_reduce<0>(kl, sred);
  float ge[8], gt[8];
  float n1 = 0.f, n2 = 0.f, n3 = 0.f, n4 = 0.f;
#pragma unroll
  for (int i = 0; i < 8; ++i) {
    float lp = ze[i] - me - lse, lq = zt[i] - mt - lst;
    float p = expf(lp), q = expf(lq);
    ge[i] = p * ((lp - lq) - kl) * invB;   // d dist / d env logits
    gt[i] = (q - p) * invB;                // d dist / d total logits
    float a = ge[i], bb = ge[i] * ze[i], cc = gt[i], dd = gt[i] * zt[i];
    n1 += a * a; n2 += bb * bb; n3 += cc * cc; n4 += dd * dd;
  }
  n1 = block_reduce<0>(n1, sred);
  n2 = block_reduce<0>(n2, sred);
  n3 = block_reduce<0>(n3, sred);
  n4 = block_reduce<0>(n4, sred);
  // perturb divides by sqrt(norm + 1e-12) where norm = sqrt(sum of squares)
  const float d1 = 1.0f / sqrtf(sqrtf(n1) + 1e-12f);
  const float d2 = 1.0f / sqrtf(sqrtf(n2) + 1e-12f);
  const float d3 = 1.0f / sqrtf(sqrtf(n3) + 1e-12f);
  const float d4 = 1.0f / sqrtf(sqrtf(n4) + 1e-12f);
#pragma unroll
  for (int i = 0; i < 8; ++i) {
    int c = tid + i * 256;
    float ve = 1.0f + rho * (ge[i] * ze[i]) * d2;  // var_e'
    float vt = 1.0f + rho * (gt[i] * zt[i]) * d4;  // var_t'
    enva[b * C_ + c] = ze[i] * ve + rho * ge[i] * d1;
    tota[b * C_ + c] = zt[i] * vt + rho * gt[i] * d3;
  }
}

// ---------------- K4: gram[c] = sum_b (tot+1e-7)(env+1e-7) via WMMA ----------
// One wave per 16 channels. A (16x4 f32, documented layout): rows = channels,
// K = 4 batch rows per step; B = all-ones (layout independent); 32 chained
// accumulations leave gram[c0+m] replicated across all N columns of D.
__global__ void k_gram(const float* __restrict__ tota, const float* __restrict__ enva,
                       float* __restrict__ gram) {
  const int c0   = blockIdx.x * 16;
  const int lane = threadIdx.x;       // blockDim = 32, full wave, EXEC all ones
  const int half = lane >> 4;         // 0: K=0,1   1: K=2,3
  const int c    = c0 + (lane & 15);
  v8f acc = {0.f, 0.f, 0.f, 0.f, 0.f, 0.f, 0.f, 0.f};
  v2f ones; ones.x = 1.0f; ones.y = 1.0f;
  for (int b0 = 0; b0 < B_; b0 += 4) {
    const int r0 = b0 + 2 * half;
    v2f a;
    a.x = (tota[r0 * C_ + c] + 1e-7f) * (enva[r0 * C_ + c] + 1e-7f);
    a.y = (tota[(r0 + 1) * C_ + c] + 1e-7f) * (enva[(r0 + 1) * C_ + c] + 1e-7f);
    acc = __builtin_amdgcn_wmma_f32_16x16x4_f32(false, a, false, ones,
                                                (short)0, acc, false, false);
  }
  // D layout: lanes 0-15 hold M=0..7 in v[0..7]; lanes 16-31 hold M=8..15
  if (lane == 0) {
#pragma unroll
    for (int rr = 0; rr < 8; ++rr) gram[c0 + rr] = acc[rr];
  } else if (lane == 16) {
#pragma unroll
    for (int rr = 0; rr < 8; ++rr) gram[c0 + 8 + rr] = acc[rr];
  }
}

// ---------------- K5: minmax chains, scores, key1, per-row top-k mask --------
__global__ void k_select(const float* __restrict__ tota, const float* __restrict__ enva,
                         const float* __restrict__ gram, const float* __restrict__ rmat,
                         const float* __restrict__ ratio_p,
                         float* __restrict__ mask, float* __restrict__ rowsum) {
  __shared__ float sred[8];
  __shared__ float keys[C_];
  const int b   = blockIdx.x;
  const int tid = threadIdx.x;
  float ta[8], ea[8];
#pragma unroll
  for (int i = 0; i < 8; ++i) {
    int c = tid + i * 256;
    float g = gram[c];
    ta[i] = tota[b * C_ + c] / g;
    ea[i] = enva[b * C_ + c] / g;
  }
  float tmin = INFINITY, tmax = -INFINITY, emin = INFINITY, emax = -INFINITY;
#pragma unroll
  for (int i = 0; i < 8; ++i) {
    tmin = fminf(tmin, ta[i]); tmax = fmaxf(tmax, ta[i]);
    emin = fminf(emin, ea[i]); emax = fmaxf(emax, ea[i]);
  }
  tmin = block_reduce<2>(tmin, sred); tmax = block_reduce<1>(tmax, sred);
  emin = block_reduce<2>(emin, sred); emax = block_reduce<1>(emax, sred);
  const float tsc = 1.0f / (tmax - tmin), esc = 1.0f / (emax - emin);
  float d[8];
  float dmin = INFINITY, dmax = -INFINITY;
#pragma unroll
  for (int i = 0; i < 8; ++i) {
    float tn = (ta[i] - tmin) * tsc;
    float en = (ea[i] - emin) * esc;
    float dd = tn - en; dd *= dd;
    d[i] = dd;
    dmin = fminf(dmin, dd); dmax = fmaxf(dmax, dd);
  }
  dmin = block_reduce<2>(dmin, sred); dmax = block_reduce<1>(dmax, sred);
  const float dsc = 1.0f / (dmax - dmin);
  float key[8];
#pragma unroll
  for (int i = 0; i < 8; ++i) {
    int c = tid + i * 256;
    float s = (d[i] - dmin) * dsc;          // scores in [0,1]
    key[i] = powf(rmat[b * C_ + c], 1.0f / s);  // s=0 -> r^inf = 0 (r<1)
    keys[c] = key[i];
  }
  __syncthreads();
  // in-LDS bitonic ascending sort of 2048 keys
  for (unsigned ks = 2; ks <= (unsigned)C_; ks <<= 1) {
    for (unsigned j = ks >> 1; j > 0; j >>= 1) {
      for (unsigned t = tid; t < (unsigned)C_; t += 256) {
        unsigned ixj = t ^ j;
        if (ixj > t) {
          bool up = ((t & ks) == 0);
          float a = keys[t], bb = keys[ixj];
          bool sw = up ? (a > bb) : (a < bb);
          if (sw) { keys[t] = bb; keys[ixj] = a; }
        }
      }
      __syncthreads();
    }
  }
  const int k = (int)(ratio_p[0] * (float)C_);   // 614
  const float thr = keys[C_ - 1 - k];            // k-th in descending order
  float cnt = 0.f;
#pragma unroll
  for (int i = 0; i < 8; ++i) {
    int c = tid + i * 256;
    float m = (key[i] > thr) ? 0.f : 1.f;        // mask = 1 - (key1 > thr)
    mask[b * C_ + c] = m;
    cnt += m;
  }
  cnt = block_reduce<0>(cnt, sred);
  if (tid == 0) rowsum[b] = cnt;
}

// ---------------- K6: global rescale factor ----------------------------------
__global__ void k_scale(const float* __restrict__ rowsum, float* __restrict__ scale) {
  __shared__ float sred[4];
  float v = rowsum[threadIdx.x];                 // blockDim = 128
  v = block_reduce<0>(v, sred);
  if (threadIdx.x == 0) scale[0] = (float)(B_ * C_) / v;
}

// ---------------- K7: out = x * mask[bc] * scale (float4 streaming) ----------
__global__ void k_apply(const float* __restrict__ x, const float* __restrict__ mask,
                        const float* __restrict__ scale, float* __restrict__ out) {
  const int i4 = blockIdx.x * 256 + threadIdx.x;     // < B*C*49
  const float4* x4 = (const float4*)x;
  float4* o4 = (float4*)out;
  if (i4 + 16384 < B_ * C_ * (HW_ / 4))
    __builtin_prefetch((const void*)(x4 + i4 + 16384), 0, 0);  // global_prefetch_b8
  const int bc = i4 / (HW_ / 4);
  const float m = mask[bc] * scale[0];
  float4 v = x4[i4];
  float4 w; w.x = v.x * m; w.y = v.y * m; w.z = v.z * m; w.w = v.w * m;
  o4[i4] = w;
}

extern "C" void kernel_launch(void* const* d_in, const int* in_sizes, int n_in,
                              void* d_out, int out_size, void* d_ws, size_t ws_size,
                              hipStream_t stream) {
  const float* x     = (const float*)d_in[0];
  const float* rmat  = (const float*)d_in[1];
  const float* ratio = (const float*)d_in[2];
  const float* rho   = (const float*)d_in[3];
  float* out = (float*)d_out;
  float* ws  = (float*)d_ws;

  float* avg    = ws;                  // B*C
  float* ze     = avg    + B_ * C_;    // B*C
  float* zt     = ze     + B_ * C_;    // B*C
  float* enva   = zt     + B_ * C_;    // B*C
  float* tota   = enva   + B_ * C_;    // B*C
  float* gram   = tota   + B_ * C_;    // C
  float* maskp  = gram   + C_;         // B*C
  float* rowsum = maskp  + B_ * C_;    // B
  float* scale  = rowsum + B_;         // 1   (total ~6.3 MB)

  k_avg   <<<B_ * C_ / 8, 256, 0, stream>>>(x, avg);
  k_stats <<<C_ / 256,    256, 0, stream>>>(avg, ze, zt);
  k_align <<<B_,          256, 0, stream>>>(ze, zt, rho, enva, tota);
  k_gram  <<<C_ / 16,      32, 0, stream>>>(tota, enva, gram);
  k_select<<<B_,          256, 0, stream>>>(tota, enva, gram, rmat, ratio, maskp, rowsum);
  k_scale <<<1,           128, 0, stream>>>(rowsum, scale);
  k_apply <<<(B_ * C_ * (HW_ / 4)) / 256, 256, 0, stream>>>(x, maskp, scale, out);
}